// Layer_12902081757551
// MI455X (gfx1250) — compile-verified
//
#include <hip/hip_runtime.h>
#include <math.h>

// Problem constants (from the reference)
#define S_TOT 2048
#define KU 8
#define NRX 16
#define NTX 64
#define DD 8

typedef float v2f __attribute__((ext_vector_type(2)));
typedef float v8f __attribute__((ext_vector_type(8)));

// CDNA5 async global->LDS mover (ASYNCcnt-tracked), if this toolchain has it.
// Probe round 2 showed its params are (int4 AS1*, int4 AS3*, imm, imm).
#if defined(__AMDGCN__) &&                                        \
    __has_builtin(__builtin_amdgcn_global_load_async_to_lds_b128) && \
    __has_builtin(__builtin_amdgcn_s_wait_asynccnt)
#define HAVE_ASYNC_LDS 1
typedef __attribute__((__vector_size__(16))) int i4;
typedef __attribute__((address_space(1))) i4 g_i4;   // global int4
typedef __attribute__((address_space(3))) i4 l_i4;   // LDS int4
#else
#define HAVE_ASYNC_LDS 0
#endif

// ---- CDNA5 WMMA f32 16x16x4 wrapper -------------------------------------
// D(16x16,f32) = A(16x4,f32) * B(4x16,f32) + C
// args: (neg_a, A, neg_b, B, c_mod, C, reuse_a, reuse_b)
__device__ __forceinline__ v8f wmma4(v2f a, v2f b, v8f c) {
  return __builtin_amdgcn_wmma_f32_16x16x4_f32(false, a, false, b, (short)0, c,
                                               false, false);
}

// ---- fragment gather/scatter per ISA 16x16x4 f32 layouts ----------------
// A 16x4: lane L: m=L&15, K pair = 2*(L>>4)  -> {a.x=K0, a.y=K0+1}
__device__ __forceinline__ v2f frag_a(const float* src, int ld, int row0,
                                      int k0, int lane) {
  int m = lane & 15;
  int kr = k0 + ((lane >> 4) << 1);
  v2f a;
  a.x = src[(row0 + m) * ld + kr];
  a.y = src[(row0 + m) * ld + kr + 1];
  return a;
}
// B 4x16 from row-major src: B[kr][n] = src[kr][col0+n]
__device__ __forceinline__ v2f frag_b(const float* src, int ld, int k0,
                                      int col0, int lane) {
  int n = lane & 15;
  int kr = k0 + ((lane >> 4) << 1);
  v2f b;
  b.x = src[kr * ld + col0 + n];
  b.y = src[(kr + 1) * ld + col0 + n];
  return b;
}
// B 4x16 where B = src^T: B[kr][n] = src[col0+n][kr]
__device__ __forceinline__ v2f frag_bT(const float* src, int ld, int k0,
                                       int col0, int lane) {
  int n = lane & 15;
  int kr = k0 + ((lane >> 4) << 1);
  v2f b;
  b.x = src[(col0 + n) * ld + kr];
  b.y = src[(col0 + n) * ld + kr + 1];
  return b;
}
// B 4x16 from a 64x8 row-major src, columns 8..15 are zero (N=8 packing)
__device__ __forceinline__ v2f frag_b8(const float* src, int k0, int lane) {
  int n = lane & 15;
  int kr = k0 + ((lane >> 4) << 1);
  v2f b;
  b.x = (n < DD) ? src[kr * DD + n] : 0.0f;
  b.y = (n < DD) ? src[(kr + 1) * DD + n] : 0.0f;
  return b;
}
// D/C 16x16: VGPR r: lanes0-15 M=r, lanes16-31 M=r+8
__device__ __forceinline__ void store_tile(float* dst, int ld, int row0,
                                           int col0, int lane, v8f d) {
  int n = lane & 15;
  int mb = (lane >> 4) << 3;
#pragma unroll
  for (int r = 0; r < 8; ++r)
    dst[(row0 + mb + r) * ld + col0 + n] = d[r];
}

// One workgroup per subcarrier s; 8 waves (wave32) <-> 8 users.
__global__ __launch_bounds__(256) void wmmse_kernel(
    const float* __restrict__ Vg, const float* __restrict__ Hg,
    const float* __restrict__ XU, const float* __restrict__ YU,
    const float* __restrict__ ZU, const float* __restrict__ OU,
    const float* __restrict__ XW, const float* __restrict__ YW,
    const float* __restrict__ ZW, const float* __restrict__ XV,
    const float* __restrict__ YV, const float* __restrict__ ZV,
    const float* __restrict__ OV, float* __restrict__ Out) {
  __shared__ float sH[KU * NRX * NTX];   // 32 kB; reused as sVn after phase 2
  __shared__ float sV[KU * NTX * DD];    // 16 kB
  __shared__ float sSsum[NTX * NTX];     // 16 kB; reused as sG2/sG13 in phase 4
  __shared__ float sB[NTX * NTX];        // 16 kB
  __shared__ float sTmp[KU * 1024];      // 32 kB; per-k: T(16x64) then HtU|HtUW
  __shared__ float sA[KU * 256];         //  8 kB
  __shared__ float sHV[KU * 128];
  __shared__ float sP2[KU * 128];
  __shared__ float sU[KU * 128];
  __shared__ float sE[KU * 64];
  __shared__ float sW[KU * 64];
  __shared__ float sScal[4];             // [0]=trV [1]=trUWU [2]=ssq

  const int tid = threadIdx.x;
  const int lane = tid & 31;
  const int wave = tid >> 5;
  const int s = blockIdx.x;

  if (tid < 4) sScal[tid] = 0.0f;

  // ---- Phase 0: stage H[s], V[s] into LDS --------------------------------
  {
    const float* hsrc = Hg + (size_t)s * (KU * NRX * NTX);
    const float* vsrc = Vg + (size_t)s * (KU * NTX * DD);
#if HAVE_ASYNC_LDS
    // Direct global->LDS b128 async copies (one 16B packet per lane),
    // tracked by ASYNCcnt; no VGPR round trip.
    for (int i = tid; i < (KU * NRX * NTX) / 4; i += 256)
      __builtin_amdgcn_global_load_async_to_lds_b128(
          (g_i4*)(hsrc + i * 4), (l_i4*)(sH + i * 4), 0, 0);
    for (int i = tid; i < (KU * NTX * DD) / 4; i += 256)
      __builtin_amdgcn_global_load_async_to_lds_b128(
          (g_i4*)(vsrc + i * 4), (l_i4*)(sV + i * 4), 0, 0);
    __builtin_amdgcn_s_wait_asynccnt(0);
#else
    for (int i = tid; i < KU * NRX * NTX; i += 256) sH[i] = hsrc[i];
    for (int i = tid; i < KU * NTX * DD; i += 256) sV[i] = vsrc[i];
#endif
  }
  __syncthreads();
  // trV = sum V^2 (from LDS)
  {
    float tv = 0.0f;
    for (int i = tid; i < KU * NTX * DD; i += 256) {
      float v = sV[i];
      tv += v * v;
    }
    atomicAdd(&sScal[0], tv);
  }
  __syncthreads();

  // ---- Phase 1: Ssum = sum_k V_k V_k^T  (64x64, 16 WMMA tiles) ----------
  for (int t = wave; t < 16; t += 8) {
    int ti = t >> 2, tj = t & 3;
    v8f acc = {};
    for (int k = 0; k < KU; ++k) {
      const float* vk = sV + k * (NTX * DD);
#pragma unroll
      for (int kk = 0; kk < DD; kk += 4) {
        v2f a = frag_a(vk, DD, ti * 16, kk, lane);
        v2f b = frag_bT(vk, DD, kk, tj * 16, lane);
        acc = wmma4(a, b, acc);
      }
    }
    store_tile(sSsum, NTX, ti * 16, tj * 16, lane, acc);
  }
  __syncthreads();

  // ---- Phase 2: per-user chain, wave w owns k=w --------------------------
  {  // T = H_k (16x64) @ Ssum (64x64), staged into sTmp[k]
    const int k = wave;
    const float* sHk = sH + k * (NRX * NTX);
    float* sTk = sTmp + k * 1024;
    for (int j = 0; j < 4; ++j) {
      v8f acc = {};
#pragma unroll
      for (int kk = 0; kk < NTX; kk += 4) {
        v2f a = frag_a(sHk, NTX, 0, kk, lane);
        v2f b = frag_b(sSsum, NTX, kk, j * 16, lane);
        acc = wmma4(a, b, acc);
      }
      store_tile(sTk, NTX, 0, j * 16, lane, acc);
    }
  }
  __syncthreads();
  {  // A = T @ H_k^T + trV*I ; HV = H_k @ V_k
    const int k = wave;
    const float* sHk = sH + k * (NRX * NTX);
    const float* sVk = sV + k * (NTX * DD);
    const float* sTk = sTmp + k * 1024;
    {
      v8f acc = {};
#pragma unroll
      for (int kk = 0; kk < NTX; kk += 4) {
        v2f a = frag_a(sTk, NTX, 0, kk, lane);
        v2f b = frag_bT(sHk, NTX, kk, 0, lane);
        acc = wmma4(a, b, acc);
      }
      float trv = sScal[0];  // P == 1.0
      int n = lane & 15, mb = (lane >> 4) << 3;
#pragma unroll
      for (int r = 0; r < 8; ++r) {
        float v = acc[r];
        if (mb + r == n) v += trv;
        sA[k * 256 + (mb + r) * 16 + n] = v;
      }
    }
    {
      v8f acc = {};
#pragma unroll
      for (int kk = 0; kk < NTX; kk += 4) {
        v2f a = frag_a(sHk, NTX, 0, kk, lane);
        v2f b = frag_b8(sVk, kk, lane);
        acc = wmma4(a, b, acc);
      }
      int n = lane & 15, mb = (lane >> 4) << 3;
      if (n < DD) {
#pragma unroll
        for (int r = 0; r < 8; ++r) sHV[k * 128 + (mb + r) * DD + n] = acc[r];
      }
    }
  }
  __syncthreads();
  {  // P2 = Y_U @ HV (16x8)  [small -> VALU]
    const int k = wave;
#pragma unroll
    for (int e = 0; e < 4; ++e) {
      int idx = lane * 4 + e, r = idx >> 3, d = idx & 7;
      float acc = 0.0f;
      for (int b = 0; b < 16; ++b)
        acc += YU[k * 256 + r * 16 + b] * sHV[k * 128 + b * 8 + d];
      sP2[k * 128 + idx] = acc;
    }
  }
  __syncthreads();
  {  // U = A+ (X_U HV) + A P2 + Z_U HV + O_U
    const int k = wave;
#pragma unroll
    for (int e = 0; e < 4; ++e) {
      int idx = lane * 4 + e, r = idx >> 3, d = idx & 7;
      float p1 = 0.0f, p3 = 0.0f, ap2 = 0.0f;
      for (int b = 0; b < 16; ++b) {
        float hv = sHV[k * 128 + b * 8 + d];
        p1 += XU[k * 256 + r * 16 + b] * hv;
        p3 += ZU[k * 256 + r * 16 + b] * hv;
        ap2 += sA[k * 256 + r * 16 + b] * sP2[k * 128 + b * 8 + d];
      }
      sU[k * 128 + idx] =
          p1 / sA[k * 256 + r * 16 + r] + ap2 + p3 + OU[k * 128 + r * 8 + d];
    }
  }
  __syncthreads();
  {  // E = I - U^T HV (8x8)
    const int k = wave;
#pragma unroll
    for (int e = 0; e < 2; ++e) {
      int idx = lane * 2 + e, a = idx >> 3, b = idx & 7;
      float acc = (a == b) ? 1.0f : 0.0f;
      for (int r = 0; r < 16; ++r)
        acc -= sU[k * 128 + r * 8 + a] * sHV[k * 128 + r * 8 + b];
      sE[k * 64 + idx] = acc;
    }
  }
  __syncthreads();
  {  // W = E+ X_W + E Y_W + Z_W (8x8)
    const int k = wave;
#pragma unroll
    for (int e = 0; e < 2; ++e) {
      int idx = lane * 2 + e, a = idx >> 3, c = idx & 7;
      float acc = XW[k * 64 + a * 8 + c] / sE[k * 64 + a * 8 + a] +
                  ZW[k * 64 + a * 8 + c];
      for (int b = 0; b < 8; ++b)
        acc += sE[k * 64 + a * 8 + b] * YW[k * 64 + b * 8 + c];
      sW[k * 64 + idx] = acc;
    }
  }
  __syncthreads();
  {  // HtU = H_k^T U (64x8) -> sTmp[k][0..511] (T is dead)
    const int k = wave;
    const float* sHk = sH + k * (NRX * NTX);
#pragma unroll
    for (int e = 0; e < 16; ++e) {
      int idx = lane * 16 + e, t = idx >> 3, d = idx & 7;
      float acc = 0.0f;
      for (int r = 0; r < 16; ++r)
        acc += sHk[r * 64 + t] * sU[k * 128 + r * 8 + d];
      sTmp[k * 1024 + idx] = acc;
    }
  }
  __syncthreads();
  {  // HtUW = HtU @ W -> sTmp[k][512..1023]; trUWU partial
    const int k = wave;
#pragma unroll
    for (int e = 0; e < 16; ++e) {
      int idx = lane * 16 + e, t = idx >> 3, d = idx & 7;
      float acc = 0.0f;
      for (int b = 0; b < 8; ++b)
        acc += sTmp[k * 1024 + t * 8 + b] * sW[k * 64 + b * 8 + d];
      sTmp[k * 1024 + 512 + idx] = acc;
    }
    float part = 0.0f;
#pragma unroll
    for (int e = 0; e < 4; ++e) {
      int idx = lane * 4 + e, r = idx >> 3, d = idx & 7;
      float uw = 0.0f;
      for (int b = 0; b < 8; ++b)
        uw += sU[k * 128 + r * 8 + b] * sW[k * 64 + b * 8 + d];
      part += uw * sU[k * 128 + idx];
    }
    atomicAdd(&sScal[1], part);
  }
  __syncthreads();

  // ---- Phase 3: B = sum_k HtUW_k HtU_k^T + trUWU*I (16 WMMA tiles) ------
  for (int t = wave; t < 16; t += 8) {
    int ti = t >> 2, tj = t & 3;
    v8f acc = {};
    for (int k = 0; k < KU; ++k) {
      const float* htuw = sTmp + k * 1024 + 512;
      const float* htu = sTmp + k * 1024;
#pragma unroll
      for (int kk = 0; kk < DD; kk += 4) {
        v2f a = frag_a(htuw, DD, ti * 16, kk, lane);
        v2f b = frag_bT(htu, DD, kk, tj * 16, lane);
        acc = wmma4(a, b, acc);
      }
    }
    float tr = sScal[1];  // P == 1.0
    int n = lane & 15, mb = (lane >> 4) << 3;
#pragma unroll
    for (int r = 0; r < 8; ++r) {
      float v = acc[r];
      if (ti == tj && mb + r == n) v += tr;
      sB[(ti * 16 + mb + r) * NTX + tj * 16 + n] = v;
    }
  }
  __syncthreads();

  // ---- Phase 4: Vn_k = B+ (X_V HtUW) + B (Y_V HtUW) + Z_V HtUW + O_V ----
  float* sG2 = sSsum;         // 64x8 scratch (Ssum dead)
  float* sG13 = sSsum + 512;  // 64x8 scratch
  float* sVn = sH;            // K*64*8 staging (H dead)
  for (int k = 0; k < KU; ++k) {
    const float* htuw = sTmp + k * 1024 + 512;
    int ti = wave & 3;
    if (wave < 4) {  // waves 0-3: G2 = Y_V @ HtUW (row tile ti)
      v8f acc = {};
#pragma unroll
      for (int kk = 0; kk < NTX; kk += 4) {
        v2f a = frag_a(YV + k * 4096, NTX, ti * 16, kk, lane);
        v2f b = frag_b8(htuw, kk, lane);
        acc = wmma4(a, b, acc);
      }
      int n = lane & 15, mb = (lane >> 4) << 3;
      if (n < DD) {
#pragma unroll
        for (int r = 0; r < 8; ++r) sG2[(ti * 16 + mb + r) * DD + n] = acc[r];
      }
    } else {  // waves 4-7: G13 = diag(1/diag B)*(X_V HtUW) + Z_V HtUW + O_V
      v8f acc1 = {}, acc3 = {};
#pragma unroll
      for (int kk = 0; kk < NTX; kk += 4) {
        v2f b = frag_b8(htuw, kk, lane);
        v2f a1 = frag_a(XV + k * 4096, NTX, ti * 16, kk, lane);
        acc1 = wmma4(a1, b, acc1);
        v2f a3 = frag_a(ZV + k * 4096, NTX, ti * 16, kk, lane);
        acc3 = wmma4(a3, b, acc3);
      }
      int n = lane & 15, mb = (lane >> 4) << 3;
      if (n < DD) {
#pragma unroll
        for (int r = 0; r < 8; ++r) {
          int m = ti * 16 + mb + r;
          sG13[m * DD + n] = acc1[r] / sB[m * NTX + m] + acc3[r] +
                             OV[k * 512 + m * 8 + n];
        }
      }
    }
    __syncthreads();
    if (wave < 4) {  // Vn tile = B @ G2 + G13
      v8f acc = {};
#pragma unroll
      for (int kk = 0; kk < NTX; kk += 4) {
        v2f a = frag_a(sB, NTX, ti * 16, kk, lane);
        v2f b = frag_b8(sG2, kk, lane);
        acc = wmma4(a, b, acc);
      }
      int n = lane & 15, mb = (lane >> 4) << 3;
      if (n < DD) {
#pragma unroll
        for (int r = 0; r < 8; ++r) {
          int m = ti * 16 + mb + r;
          sVn[k * 512 + m * DD + n] = acc[r] + sG13[m * DD + n];
        }
      }
    }
    __syncthreads();
  }

  // ---- Phase 5: power projection and write-out ---------------------------
  {
    float part = 0.0f;
    for (int i = tid; i < KU * NTX * DD; i += 256) {
      float v = sVn[i];
      part += v * v;
    }
    atomicAdd(&sScal[2], part);
  }
  __syncthreads();
  {
    float scale = sqrtf(1.0f / sScal[2]);  // P == 1.0
    float* dst = Out + (size_t)s * (KU * NTX * DD);
    for (int i = tid; i < KU * NTX * DD; i += 256) dst[i] = sVn[i] * scale;
  }
}

extern "C" void kernel_launch(void* const* d_in, const int* in_sizes, int n_in,
                              void* d_out, int out_size, void* d_ws,
                              size_t ws_size, hipStream_t stream) {
  (void)in_sizes; (void)n_in; (void)out_size; (void)d_ws; (void)ws_size;
  const float* Vg = (const float*)d_in[0];
  const float* Hg = (const float*)d_in[1];
  const float* XU = (const float*)d_in[2];
  const float* YU = (const float*)d_in[3];
  const float* ZU = (const float*)d_in[4];
  const float* OU = (const float*)d_in[5];
  const float* XW = (const float*)d_in[6];
  const float* YW = (const float*)d_in[7];
  const float* ZW = (const float*)d_in[8];
  const float* XV = (const float*)d_in[9];
  const float* YV = (const float*)d_in[10];
  const float* ZV = (const float*)d_in[11];
  const float* OV = (const float*)d_in[12];
  wmmse_kernel<<<dim3(S_TOT), dim3(256), 0, stream>>>(
      Vg, Hg, XU, YU, ZU, OU, XW, YW, ZW, XV, YV, ZV, OV, (float*)d_out);
}